// YoloLoss_49426483642483
// MI455X (gfx1250) — compile-verified
//
#include <hip/hip_runtime.h>
#include <hip/hip_bf16.h>

// ---------------------------------------------------------------------------
// YOLO loss, MI455X (gfx1250, wave32).
// Memory-bound streaming reduction: 192.7 MB in -> 1 float out
// (roofline: ~8.3 us at 23.3 TB/s; ~0.16 GFLOP of VALU -> pure streaming).
// Each thread owns TWO cells (240 B = 15 x float4, 16B-aligned) so all
// global traffic is global_load_b128, 512 B per wave per instruction.
// Reduction: exact f32 wave sum via V_WMMA_F32_16X16X4_F32 (A = partials,
// B = ones -> D[m][n] = x_m + x_{m+16}), LDS across 8 waves, then a
// 1-block kernel reduces the 1568 block partials (again via WMMA).
// No atomics -> bitwise deterministic across graph replays.
// ---------------------------------------------------------------------------

typedef __attribute__((ext_vector_type(2))) float v2f;
typedef __attribute__((ext_vector_type(8))) float v8f;

#define NCELLS (4096 * 196)        // BATCH * S2 = 802816
#define NPAIRS (NCELLS / 2)        // 401408
#define NBLOCKS (NPAIRS / 256)     // 1568, exact

// Exact f32 sum of all 32 lanes of the wave; result broadcast to all lanes.
// A (16x4): lanes0-15 VGPR0=K0, lanes16-31 VGPR0=K2 (VGPR1 -> K1/K3 = 0);
// B = ones. D[m][n] = x_m + x_{m+16}; 8 row-sums per lane + one xor-16 add.
__device__ __forceinline__ float wave_sum_wmma(float x) {
  v2f a; a.x = x;    a.y = 0.0f;
  v2f b; b.x = 1.0f; b.y = 1.0f;
  v8f c = {};
  c = __builtin_amdgcn_wmma_f32_16x16x4_f32(
      /*neg_a=*/false, a, /*neg_b=*/false, b,
      /*c_mod=*/(short)0, c, /*reuse_a=*/false, /*reuse_b=*/false);
  float s = ((c[0] + c[1]) + (c[2] + c[3])) + ((c[4] + c[5]) + (c[6] + c[7]));
  s += __shfl_xor(s, 16, 32);
  return s;
}

// Loss for a single grid cell; pv/tv point at 30 floats each (register arrays).
__device__ __forceinline__ float cell_loss(const float* pv, const float* tv) {
  const float t5    = tv[5];
  const float coord = (t5 > 0.0f)  ? 1.0f : 0.0f;
  const float noobj = (t5 == 0.0f) ? 1.0f : 0.0f;

  // class loss over channels 10..29
  float cls = 0.0f;
#pragma unroll
  for (int e = 10; e < 30; ++e) {
    float d = pv[e] - tv[e];
    cls = fmaf(d, d, cls);
  }

  // confidence diffs (indices 4, 9)
  const float d4 = pv[4] - tv[4];
  const float d9 = pv[9] - tv[9];
  const float noo = d4 * d4 + d9 * d9;

  // box corners + areas (mirror reference arithmetic order)
  float c1x[2], c1y[2], c1z[2], c1w[2], a1[2];
  float c2x[2], c2y[2], c2z[2], c2w[2], a2[2];
#pragma unroll
  for (int i = 0; i < 2; ++i) {
    float x = pv[5 * i], y = pv[5 * i + 1], wv = pv[5 * i + 2], hv = pv[5 * i + 3];
    float w2 = wv * wv, h2 = hv * hv;
    c1x[i] = x - w2; c1y[i] = y - h2; c1z[i] = x + w2; c1w[i] = y + h2;
    a1[i] = (c1z[i] - c1x[i]) * (c1w[i] - c1y[i]);

    x = tv[5 * i]; y = tv[5 * i + 1]; wv = tv[5 * i + 2]; hv = tv[5 * i + 3];
    w2 = wv * wv; h2 = hv * hv;
    c2x[i] = x - w2; c2y[i] = y - h2; c2z[i] = x + w2; c2w[i] = y + h2;
    a2[i] = (c2z[i] - c2x[i]) * (c2w[i] - c2y[i]);
  }

  // 2x2 IOU table: i = pred box, j = target box
  float iou[2][2];
#pragma unroll
  for (int i = 0; i < 2; ++i) {
#pragma unroll
    for (int j = 0; j < 2; ++j) {
      float tlx = fmaxf(c1x[i], c2x[j]);
      float tly = fmaxf(c1y[i], c2y[j]);
      float brx = fminf(c1z[i], c2z[j]);
      float bry = fminf(c1w[i], c2w[j]);
      float wx = fmaxf(brx - tlx, 0.0f);
      float wy = fmaxf(bry - tly, 0.0f);
      float inter = wx * wy;
      float un = a1[i] + a2[j] - inter;
      iou[i][j] = inter / fmaxf(un, 1e-12f);
    }
  }

  // argmax over pred boxes per target box (first-max wins ties, like jnp)
  const int idx0 = (iou[1][0] > iou[0][0]) ? 1 : 0;
  const int idx1 = (iou[1][1] > iou[0][1]) ? 1 : 0;
  const float resp0 = (idx0 == 0 || idx1 == 0) ? 1.0f : 0.0f;
  const float resp1 = (idx0 == 1 || idx1 == 1) ? 1.0f : 0.0f;
  const float w0 = coord * resp0;
  const float w1 = coord * resp1;

  const float contain = w0 * d4 * d4 + w1 * d9 * d9;

  float loc = 0.0f;
#pragma unroll
  for (int k = 0; k < 4; ++k) {
    float dp0 = pv[k] - tv[k];
    float dp1 = pv[5 + k] - tv[5 + k];
    loc = fmaf(w0 * dp0, dp0, loc);
    loc = fmaf(w1 * dp1, dp1, loc);
  }

  return 5.0f * loc + contain + 0.5f * noobj * noo + coord * cls;
}

__global__ void __launch_bounds__(256)
yolo_loss_main(const float* __restrict__ pred,
               const float* __restrict__ tgt,
               float* __restrict__ blocksums) {
  const int pair = blockIdx.x * 256 + threadIdx.x;

  float loss = 0.0f;
  if (pair < NPAIRS) {
    // Two consecutive cells = 240 bytes = 15 float4, 16B-aligned.
    float pv[60], tv[60];
    const float4* p4 = reinterpret_cast<const float4*>(pred) + (size_t)pair * 15;
    const float4* t4 = reinterpret_cast<const float4*>(tgt)  + (size_t)pair * 15;
#pragma unroll
    for (int i = 0; i < 15; ++i) {
      float4 pa = p4[i];
      pv[4 * i + 0] = pa.x; pv[4 * i + 1] = pa.y;
      pv[4 * i + 2] = pa.z; pv[4 * i + 3] = pa.w;
      float4 ta = t4[i];
      tv[4 * i + 0] = ta.x; tv[4 * i + 1] = ta.y;
      tv[4 * i + 2] = ta.z; tv[4 * i + 3] = ta.w;
    }
    loss = cell_loss(pv, tv) + cell_loss(pv + 30, tv + 30);
  }

  // --- block reduction: WMMA wave sum, then 8 wave partials via LDS ---
  __shared__ float wsum[8];
  const float wtot = wave_sum_wmma(loss);   // EXEC all-ones (uniform path)
  const int lane = threadIdx.x & 31;
  const int wid  = threadIdx.x >> 5;
  if (lane == 0) wsum[wid] = wtot;
  __syncthreads();
  if (threadIdx.x == 0) {
    float s = 0.0f;
#pragma unroll
    for (int i = 0; i < 8; ++i) s += wsum[i];
    blocksums[blockIdx.x] = s;
  }
}

__global__ void __launch_bounds__(256)
yolo_loss_final(const float* __restrict__ blocksums,
                float* __restrict__ out) {
  float acc = 0.0f;
  for (int i = threadIdx.x; i < NBLOCKS; i += 256) acc += blocksums[i];

  __shared__ float wsum[8];
  const float wtot = wave_sum_wmma(acc);
  const int lane = threadIdx.x & 31;
  const int wid  = threadIdx.x >> 5;
  if (lane == 0) wsum[wid] = wtot;
  __syncthreads();
  if (threadIdx.x == 0) {
    float s = 0.0f;
#pragma unroll
    for (int i = 0; i < 8; ++i) s += wsum[i];
    out[0] = s;
  }
}

extern "C" void kernel_launch(void* const* d_in, const int* in_sizes, int n_in,
                              void* d_out, int out_size, void* d_ws, size_t ws_size,
                              hipStream_t stream) {
  const float* pred = (const float*)d_in[0];   // (4096, 196*30) f32
  const float* tgt  = (const float*)d_in[1];   // (4096, 14, 14, 30) f32
  float* blocksums  = (float*)d_ws;            // NBLOCKS floats of scratch
  float* out        = (float*)d_out;           // 1 float

  yolo_loss_main<<<NBLOCKS, 256, 0, stream>>>(pred, tgt, blocksums);
  yolo_loss_final<<<1, 256, 0, stream>>>(blocksums, out);
}